// PoseODERNN_3401614098640
// MI455X (gfx1250) — compile-verified
//
#include <hip/hip_runtime.h>
#include <hip/hip_bf16.h>
#include <math.h>

// ---------------------------------------------------------------------------
// PoseODERNN on MI455X (gfx1250).
// All GEMMs use V_WMMA_F32_16X16X4_F32 (full-fp32 matrix pipeline) so the
// adaptive DOPRI5 accept/reject decisions match the f32 reference.
// Per-wave tile is 64x32 (8 WMMA accumulators) to amortize the K=4 operand
// traffic: 8 WMMAs per 4xB64 + 4xB32 loads (~10.7 FLOP/byte from cache).
// ---------------------------------------------------------------------------

typedef __attribute__((ext_vector_type(2))) float v2f;
typedef __attribute__((ext_vector_type(8))) float v8f;

#define B_SZ   512
#define F_SZ   768
#define H_SZ   512
#define NELEM  (B_SZ * F_SZ)   // 393216
#define NPART  192
#define ATOL_C 1e-6f
#define RTOL_C 1e-3f

// ---------------- WMMA GEMM: C = act(A[M,K] @ W[K,N] + bias[N] (+ add)) ----
// Per-wave tile: 64 rows x 32 cols = 4 M-subtiles x 2 N-subtiles.
// f32 16x16x4 operand layouts (ISA 7.12.2):
//   A 16x4 : lanes 0-15 -> M=lane, {v0,v1}={K=k,k+1}; lanes 16-31 -> {k+2,k+3}
//   B 4x16 : lanes 0-15 -> N=lane, {v0,v1}={K=k,k+1}; lanes 16-31 -> {k+2,k+3}
//   C 16x16: VGPR r, lanes 0-15 -> M=r, N=lane; lanes 16-31 -> M=8+r
// ACT: 0=none, 1=tanh, 2=leaky-relu(0.1)
template <int ACT, bool HAS_ADD>
__global__ __launch_bounds__(256) void gemm_wmma_f32(
    const float* __restrict__ A, const float* __restrict__ W,
    const float* __restrict__ bias, const float* __restrict__ addsrc,
    float* __restrict__ C, int M, int N, int K)
{
    const int lane   = threadIdx.x & 31;
    const int wave   = threadIdx.x >> 5;
    const int tilesN = N >> 5;                          // 32-wide N strips
    const int tile   = blockIdx.x * (blockDim.x >> 5) + wave;
    const int nTiles = (M >> 6) * tilesN;
    if (tile >= nTiles) return;          // whole-wave uniform exit

    const int m0   = (tile / tilesN) << 6;
    const int n0   = (tile % tilesN) << 5;
    const int half = lane >> 4;          // 0: lanes 0-15, 1: lanes 16-31
    const int l    = lane & 15;

    v8f acc[8];
    #pragma unroll
    for (int i = 0; i < 8; ++i) acc[i] = {};

    const size_t rK0 = (size_t)(m0 +  0 + l) * K;
    const size_t rK1 = (size_t)(m0 + 16 + l) * K;
    const size_t rK2 = (size_t)(m0 + 32 + l) * K;
    const size_t rK3 = (size_t)(m0 + 48 + l) * K;
    const int    col0 = n0 + l;
    const int    col1 = n0 + 16 + l;

    for (int k = 0; k < K; k += 4) {
        const int ka = k + 2 * half;
        v2f bv0, bv1;
        bv0.x = W[(size_t)(ka + 0) * N + col0];
        bv0.y = W[(size_t)(ka + 1) * N + col0];
        bv1.x = W[(size_t)(ka + 0) * N + col1];
        bv1.y = W[(size_t)(ka + 1) * N + col1];
        v2f a0 = *(const v2f*)(A + rK0 + ka);   // 8B aligned: K%4==0, ka even
        v2f a1 = *(const v2f*)(A + rK1 + ka);
        v2f a2 = *(const v2f*)(A + rK2 + ka);
        v2f a3 = *(const v2f*)(A + rK3 + ka);
        acc[0] = __builtin_amdgcn_wmma_f32_16x16x4_f32(false, a0, false, bv0, (short)0, acc[0], false, false);
        acc[1] = __builtin_amdgcn_wmma_f32_16x16x4_f32(false, a1, false, bv0, (short)0, acc[1], false, false);
        acc[2] = __builtin_amdgcn_wmma_f32_16x16x4_f32(false, a2, false, bv0, (short)0, acc[2], false, false);
        acc[3] = __builtin_amdgcn_wmma_f32_16x16x4_f32(false, a3, false, bv0, (short)0, acc[3], false, false);
        acc[4] = __builtin_amdgcn_wmma_f32_16x16x4_f32(false, a0, false, bv1, (short)0, acc[4], false, false);
        acc[5] = __builtin_amdgcn_wmma_f32_16x16x4_f32(false, a1, false, bv1, (short)0, acc[5], false, false);
        acc[6] = __builtin_amdgcn_wmma_f32_16x16x4_f32(false, a2, false, bv1, (short)0, acc[6], false, false);
        acc[7] = __builtin_amdgcn_wmma_f32_16x16x4_f32(false, a3, false, bv1, (short)0, acc[7], false, false);
    }

    const float bn0 = bias[col0];
    const float bn1 = bias[col1];
    #pragma unroll
    for (int h = 0; h < 2; ++h) {                     // N-subtile
        const int   col = h ? col1 : col0;
        const float bn  = h ? bn1  : bn0;
        #pragma unroll
        for (int s = 0; s < 4; ++s) {                 // M-subtile
            #pragma unroll
            for (int r = 0; r < 8; ++r) {
                const int row = m0 + s * 16 + half * 8 + r;
                const size_t idx = (size_t)row * N + col;
                float v = acc[h * 4 + s][r] + bn;
                if (HAS_ADD) v += addsrc[idx];
                if (ACT == 1)      v = tanhf(v);
                else if (ACT == 2) v = (v > 0.f) ? v : 0.1f * v;
                C[idx] = v;
            }
        }
    }
}

// ---------------- init: y0 = 0, control block ------------------------------
// ctl layout: [0]=t, [1]=dt, [2]=dt_c, [3]=done flag, [4]=accept flag
__global__ __launch_bounds__(256) void init_kernel(float* __restrict__ y,
                                                   float* __restrict__ ctl,
                                                   const float* __restrict__ ts, int n)
{
    const int i = blockIdx.x * blockDim.x + threadIdx.x;
    if (i < n) y[i] = 0.f;
    if (i == 0) {
        ctl[0] = ts[0];
        ctl[1] = (ts[1] - ts[0]) * 0.1f;  // dt0
        ctl[2] = 0.f; ctl[3] = 0.f; ctl[4] = 0.f;
    }
}

__global__ __launch_bounds__(256) void build_fused(const float* __restrict__ fv,
                                                   const float* __restrict__ fi,
                                                   float* __restrict__ fused)
{
    const int i = blockIdx.x * blockDim.x + threadIdx.x;
    if (i >= NELEM) return;
    const int b = i / F_SZ, c = i % F_SZ;
    fused[i] = (c < 512) ? fv[b * 512 + c] : fi[b * 256 + (c - 512)];
}

// ---------------- per-step control ----------------------------------------
__global__ void step_begin(float* __restrict__ ctl, const float* __restrict__ ts)
{
    const float t  = ctl[0];
    const float dt = ctl[1];
    const float t1 = ts[1];
    float rem = t1 - t;
    if (rem < 1e-10f) rem = 1e-10f;
    ctl[2] = (dt < rem) ? dt : rem;               // dt_c
    ctl[3] = (t >= t1 - 1e-10f) ? 1.f : 0.f;      // done
}

// X = y + dt_c * (c1*p1 + ... up to nk terms)
__global__ __launch_bounds__(256) void combine_stage(
    float* __restrict__ X, const float* __restrict__ y,
    const float* __restrict__ p1, const float* __restrict__ p2,
    const float* __restrict__ p3, const float* __restrict__ p4,
    const float* __restrict__ p5,
    float c1, float c2, float c3, float c4, float c5, int nk,
    const float* __restrict__ ctl, int n)
{
    const int i = blockIdx.x * blockDim.x + threadIdx.x;
    if (i >= n) return;
    const float dtc = ctl[2];
    float s = c1 * p1[i];
    if (nk > 1) s += c2 * p2[i];
    if (nk > 2) s += c3 * p3[i];
    if (nk > 3) s += c4 * p4[i];
    if (nk > 4) s += c5 * p5[i];
    X[i] = y[i] + dtc * s;
}

// sum of (err/tol)^2 -> fixed per-block partial slots (deterministic)
__global__ __launch_bounds__(256) void err_reduce(
    const float* __restrict__ y, const float* __restrict__ ynew,
    const float* __restrict__ k1, const float* __restrict__ k3,
    const float* __restrict__ k4, const float* __restrict__ k5,
    const float* __restrict__ k6, const float* __restrict__ k7,
    const float* __restrict__ ctl, float* __restrict__ partials, int n)
{
    const float E1 = 71.f / 57600.f,   E3 = -71.f / 16695.f, E4 = 71.f / 1920.f;
    const float E5 = -17253.f / 339200.f, E6 = 22.f / 525.f, E7 = -1.f / 40.f;
    const float dtc = ctl[2];
    float local = 0.f;
    for (int i = blockIdx.x * blockDim.x + threadIdx.x; i < n;
         i += gridDim.x * blockDim.x) {
        const float err = dtc * (E1 * k1[i] + E3 * k3[i] + E4 * k4[i] +
                                 E5 * k5[i] + E6 * k6[i] + E7 * k7[i]);
        const float tol = ATOL_C + RTOL_C * fmaxf(fabsf(y[i]), fabsf(ynew[i]));
        const float r = err / tol;
        local += r * r;
    }
    __shared__ float sm[256];
    sm[threadIdx.x] = local;
    __syncthreads();
    for (int s = 128; s > 0; s >>= 1) {
        if ((int)threadIdx.x < s) sm[threadIdx.x] += sm[threadIdx.x + s];
        __syncthreads();
    }
    if (threadIdx.x == 0) partials[blockIdx.x] = sm[0];
}

__global__ void control_update(float* __restrict__ ctl,
                               const float* __restrict__ partials,
                               int npart, int n)
{
    float sum = 0.f;
    for (int i = 0; i < npart; ++i) sum += partials[i];   // fixed order
    const float norm = sqrtf(sum / (float)n);
    const float dtc  = ctl[2];
    const bool  done = (ctl[3] != 0.f);
    const bool  accept = (norm <= 1.f) && !done;
    if (accept) ctl[0] = ctl[0] + dtc;
    const float nrm = (norm > 1e-10f) ? norm : 1e-10f;
    float factor = 0.9f * powf(nrm, -0.2f);
    factor = fminf(fmaxf(factor, 0.2f), 10.f);
    if (!done) ctl[1] = dtc * factor;
    ctl[4] = accept ? 1.f : 0.f;
}

__global__ __launch_bounds__(256) void y_update(float* __restrict__ y,
                                                const float* __restrict__ ynew,
                                                const float* __restrict__ ctl, int n)
{
    const int i = blockIdx.x * blockDim.x + threadIdx.x;
    if (i >= n) return;
    if (ctl[4] != 0.f) y[i] = ynew[i];
}

// tiny N=6 final GEMM: pose = z3 @ w4 + b4  -> d_out[0:3072]
__global__ __launch_bounds__(256) void pose_kernel(const float* __restrict__ z3,
                                                   const float* __restrict__ w4,
                                                   const float* __restrict__ b4,
                                                   float* __restrict__ out)
{
    const int idx = blockIdx.x * blockDim.x + threadIdx.x;
    if (idx >= B_SZ * 6) return;
    const int b = idx / 6, j = idx % 6;
    float s = b4[j];
    #pragma unroll 4
    for (int k = 0; k < 128; ++k) s += z3[b * 128 + k] * w4[k * 6 + j];
    out[idx] = s;
}

// ---------------- host-side launch helpers ---------------------------------
template <int ACT, bool HAS_ADD>
static inline void launch_gemm(const float* A, const float* W, const float* bias,
                               const float* addsrc, float* C, int M, int N, int K,
                               hipStream_t stream)
{
    const int tiles  = (M / 64) * (N / 32);
    const int blocks = (tiles + 7) / 8;                   // 8 waves per block
    gemm_wmma_f32<ACT, HAS_ADD><<<blocks, 256, 0, stream>>>(A, W, bias, addsrc, C, M, N, K);
}

extern "C" void kernel_launch(void* const* d_in, const int* in_sizes, int n_in,
                              void* d_out, int out_size, void* d_ws, size_t ws_size,
                              hipStream_t stream)
{
    (void)in_sizes; (void)n_in; (void)out_size; (void)ws_size;

    const float* fv        = (const float*)d_in[0];
    const float* fi        = (const float*)d_in[2];
    const float* ts        = (const float*)d_in[4];
    const float* ode_w_in  = (const float*)d_in[5];
    const float* ode_b_in  = (const float*)d_in[6];
    const float* ode_w_h   = (const float*)d_in[7];
    const float* ode_b_h   = (const float*)d_in[8];
    const float* ode_w_out = (const float*)d_in[9];
    const float* ode_b_out = (const float*)d_in[10];
    const float* rnn_w_ih  = (const float*)d_in[11];
    const float* rnn_w_hh  = (const float*)d_in[12];
    const float* rnn_b_ih  = (const float*)d_in[13];
    const float* rnn_b_hh  = (const float*)d_in[14];
    const float* reg_w1    = (const float*)d_in[15];
    const float* reg_b1    = (const float*)d_in[16];
    const float* reg_w2    = (const float*)d_in[17];
    const float* reg_b2    = (const float*)d_in[18];
    const float* reg_w3    = (const float*)d_in[19];
    const float* reg_b3    = (const float*)d_in[20];
    const float* reg_w4    = (const float*)d_in[21];
    const float* reg_b4    = (const float*)d_in[22];
    float* out = (float*)d_out;

    // workspace carve-up (64-float aligned slices)
    float* ws = (float*)d_ws;
    size_t off = 0;
    auto alloc = [&](size_t n) { float* p = ws + off; off += (n + 63) & ~(size_t)63; return p; };
    float* y    = alloc(NELEM);
    float* ynew = alloc(NELEM);
    float* X    = alloc(NELEM);
    float* k1   = alloc(NELEM);
    float* k2   = alloc(NELEM);
    float* k3   = alloc(NELEM);
    float* k4   = alloc(NELEM);
    float* k5   = alloc(NELEM);
    float* k6   = alloc(NELEM);
    float* k7   = alloc(NELEM);
    float* h1   = alloc((size_t)B_SZ * H_SZ);
    float* h2   = alloc((size_t)B_SZ * H_SZ);
    float* fused= alloc(NELEM);
    float* g1   = alloc(NELEM);
    float* z1   = alloc((size_t)B_SZ * 128);
    float* z2   = alloc((size_t)B_SZ * 256);
    float* z3   = alloc((size_t)B_SZ * 128);
    float* part = alloc(NPART);
    float* ctl  = alloc(16);

    const int EB = NELEM / 256;   // 1536 elementwise blocks

    // ODE RHS: kout = f(in)  (768 -> 512 tanh -> 512 tanh -> 768)
    auto rhs = [&](const float* in, float* kout) {
        launch_gemm<1, false>(in, ode_w_in, ode_b_in, nullptr, h1, B_SZ, H_SZ, F_SZ, stream);
        launch_gemm<1, false>(h1, ode_w_h,  ode_b_h,  nullptr, h2, B_SZ, H_SZ, H_SZ, stream);
        launch_gemm<0, false>(h2, ode_w_out, ode_b_out, nullptr, kout, B_SZ, F_SZ, H_SZ, stream);
    };

    // Dormand-Prince 5(4) tableau
    const float a21 = 1.f/5.f;
    const float a31 = 3.f/40.f, a32 = 9.f/40.f;
    const float a41 = 44.f/45.f, a42 = -56.f/15.f, a43 = 32.f/9.f;
    const float a51 = 19372.f/6561.f, a52 = -25360.f/2187.f, a53 = 64448.f/6561.f, a54 = -212.f/729.f;
    const float a61 = 9017.f/3168.f, a62 = -355.f/33.f, a63 = 46732.f/5247.f, a64 = 49.f/176.f, a65 = -5103.f/18656.f;
    const float b1 = 35.f/384.f, b3 = 500.f/1113.f, b4 = 125.f/192.f, b5 = -2187.f/6784.f, b6 = 11.f/84.f;

    init_kernel<<<EB, 256, 0, stream>>>(y, ctl, ts, NELEM);
    build_fused<<<EB, 256, 0, stream>>>(fv, fi, fused);

    for (int step = 0; step < 24; ++step) {
        step_begin<<<1, 1, 0, stream>>>(ctl, ts);
        rhs(y, k1);
        combine_stage<<<EB, 256, 0, stream>>>(X, y, k1, k1, k1, k1, k1,
                                              a21, 0.f, 0.f, 0.f, 0.f, 1, ctl, NELEM);
        rhs(X, k2);
        combine_stage<<<EB, 256, 0, stream>>>(X, y, k1, k2, k1, k1, k1,
                                              a31, a32, 0.f, 0.f, 0.f, 2, ctl, NELEM);
        rhs(X, k3);
        combine_stage<<<EB, 256, 0, stream>>>(X, y, k1, k2, k3, k1, k1,
                                              a41, a42, a43, 0.f, 0.f, 3, ctl, NELEM);
        rhs(X, k4);
        combine_stage<<<EB, 256, 0, stream>>>(X, y, k1, k2, k3, k4, k1,
                                              a51, a52, a53, a54, 0.f, 4, ctl, NELEM);
        rhs(X, k5);
        combine_stage<<<EB, 256, 0, stream>>>(X, y, k1, k2, k3, k4, k5,
                                              a61, a62, a63, a64, a65, 5, ctl, NELEM);
        rhs(X, k6);
        combine_stage<<<EB, 256, 0, stream>>>(ynew, y, k1, k3, k4, k5, k6,
                                              b1, b3, b4, b5, b6, 5, ctl, NELEM);
        rhs(ynew, k7);  // FSAL stage (error estimate only)
        err_reduce<<<NPART, 256, 0, stream>>>(y, ynew, k1, k3, k4, k5, k6, k7,
                                              ctl, part, NELEM);
        control_update<<<1, 1, 0, stream>>>(ctl, part, NPART, NELEM);
        y_update<<<EB, 256, 0, stream>>>(y, ynew, ctl, NELEM);
    }

    // RNN cell: h = tanh(fused@W_ih + b_ih + y@W_hh + b_hh) -> d_out[3072:]
    float* hbuf = out + (size_t)B_SZ * 6;
    launch_gemm<0, false>(fused, rnn_w_ih, rnn_b_ih, nullptr, g1, B_SZ, F_SZ, F_SZ, stream);
    launch_gemm<1, true >(y,     rnn_w_hh, rnn_b_hh, g1, hbuf, B_SZ, F_SZ, F_SZ, stream);

    // Regressor 768 -> 128 -> 256 -> 128 -> 6 (leaky relu 0.1)
    launch_gemm<2, false>(hbuf, reg_w1, reg_b1, nullptr, z1, B_SZ, 128, F_SZ, stream);
    launch_gemm<2, false>(z1,   reg_w2, reg_b2, nullptr, z2, B_SZ, 256, 128, stream);
    launch_gemm<2, false>(z2,   reg_w3, reg_b3, nullptr, z3, B_SZ, 128, 256, stream);
    pose_kernel<<<(B_SZ * 6 + 255) / 256, 256, 0, stream>>>(z3, reg_w4, reg_b4, out);
}